// GeodesicInterpPreimage_8959301779818
// MI455X (gfx1250) — compile-verified
//
#include <hip/hip_runtime.h>
#include <hip/hip_bf16.h>
#include <float.h>

#define B_ROWS 4096
#define N_COLS 16384
#define DIM    512
#define TOPK   16

typedef __attribute__((ext_vector_type(2))) float v2f;
typedef __attribute__((ext_vector_type(8))) float v8f;
typedef __attribute__((ext_vector_type(4))) int   v4i;

#if defined(__gfx1250__)
  #if __has_builtin(__builtin_amdgcn_global_load_async_to_lds_b128) && \
      __has_builtin(__builtin_amdgcn_s_wait_asynccnt)
    #define USE_ASYNC_LDS 1
  #endif
  #if __has_builtin(__builtin_amdgcn_wmma_f32_16x16x4_f32)
    #define USE_WMMA_F32 1
  #endif
#endif

#define GLOBAL_AS __attribute__((address_space(1)))
#define LDS_AS    __attribute__((address_space(3)))

// ---------------------------------------------------------------------------
// Kernel 1: per-row top-16 (values + indices), JAX lax.top_k ordering
// (descending value, ties broken by smaller index).
// One 256-thread block per row; each thread owns 64 elements, keeps a private
// top-16 fully in VGPRs, then 16 extraction passes merge across the block.
// ---------------------------------------------------------------------------
__global__ __launch_bounds__(256) void topk16_kernel(
    const float* __restrict__ K, int* __restrict__ out_inds,
    float* __restrict__ out_w)
{
  const int row = blockIdx.x;
  const int tid = threadIdx.x;
  const float* __restrict__ kr = K + (size_t)row * N_COLS;

  float tv[TOPK];
  int   ti[TOPK];
#pragma unroll
  for (int j = 0; j < TOPK; ++j) { tv[j] = -FLT_MAX; ti[j] = 0x7FFFFFFF; }
  float vmin = -FLT_MAX;

  // Stream the row once, 16 x float4 per thread, coalesced across threads.
#pragma unroll 1
  for (int jj = 0; jj < 16; ++jj) {
    const int base = (jj * 256 + tid) * 4;
    const float4 q = *(const float4*)(kr + base);
    const float elem[4] = {q.x, q.y, q.z, q.w};
#pragma unroll
    for (int c = 0; c < 4; ++c) {
      const float v = elem[c];
      if (v > vmin) {
        const int e = base + c;
        // Evict the current-min entry with the LARGEST original index
        // (preserves stable tie behavior at the cutoff).
        int ev = 0, evti = -1;
#pragma unroll
        for (int j = 0; j < TOPK; ++j)
          if (tv[j] == vmin && ti[j] > evti) { evti = ti[j]; ev = j; }
#pragma unroll
        for (int j = 0; j < TOPK; ++j)
          if (j == ev) { tv[j] = v; ti[j] = e; }
        vmin = tv[0];
#pragma unroll
        for (int j = 1; j < TOPK; ++j) vmin = fminf(vmin, tv[j]);
      }
    }
  }

  __shared__ float lv[8];
  __shared__ int   li[8];
  __shared__ int   lp[8];
  __shared__ float resv[TOPK];
  __shared__ int   resi[TOPK];
  __shared__ int   bpos;

#pragma unroll 1
  for (int pass = 0; pass < TOPK; ++pass) {
    // Local best among this thread's 16 candidates (tie-break lower index).
    float bv = -FLT_MAX; int boi = 0x7FFFFFFF; int bj = 0;
#pragma unroll
    for (int j = 0; j < TOPK; ++j) {
      const bool bet = (tv[j] > bv) || (tv[j] == bv && ti[j] < boi);
      if (bet) { bv = tv[j]; boi = ti[j]; bj = j; }
    }
    float v = bv; int oi = boi; int pos = tid * TOPK + bj;

    // wave32 xor-shuffle reduction
#pragma unroll
    for (int off = 16; off >= 1; off >>= 1) {
      const float ov = __shfl_xor(v, off, 32);
      const int   ooi = __shfl_xor(oi, off, 32);
      const int   op  = __shfl_xor(pos, off, 32);
      if (ov > v || (ov == v && ooi < oi)) { v = ov; oi = ooi; pos = op; }
    }
    if ((tid & 31) == 0) { const int w = tid >> 5; lv[w] = v; li[w] = oi; lp[w] = pos; }
    __syncthreads();
    if (tid == 0) {
      float fv = lv[0]; int foi = li[0]; int fp = lp[0];
#pragma unroll
      for (int w = 1; w < 8; ++w)
        if (lv[w] > fv || (lv[w] == fv && li[w] < foi)) { fv = lv[w]; foi = li[w]; fp = lp[w]; }
      resv[pass] = fv; resi[pass] = foi; bpos = fp;
    }
    __syncthreads();
    const int wp = bpos;
    if ((wp >> 4) == tid) {
      const int mj = wp & 15;
#pragma unroll
      for (int j = 0; j < TOPK; ++j)
        if (j == mj) { tv[j] = -FLT_MAX; ti[j] = 0x7FFFFFFF; }
    }
  }
  __syncthreads();
  if (tid < TOPK) {
    out_inds[(size_t)row * TOPK + tid] = resi[tid];
    out_w[(size_t)row * TOPK + tid]    = resv[tid];
  }
}

// ---------------------------------------------------------------------------
// Kernel 2: batched slerp. 16 rows per 256-thread block.
// mu/v tiles in LDS; neighbor rows gathered via async global->LDS copies;
// per-step dots via V_WMMA_F32_16X16X4_F32 (diag of mu * v^T) on wave 0.
// ---------------------------------------------------------------------------
__device__ __forceinline__ void loadrows16(const float* __restrict__ y,
                                           const int* sind, int s,
                                           float (*dst)[DIM], int tid)
{
  // 16 rows x 512 floats = 2048 float4s; 8 per thread.
#pragma unroll
  for (int j = 0; j < 8; ++j) {
    const int f = tid + 256 * j;        // 0..2047
    const int r = f >> 7;               // row 0..15
    const int q = (f & 127) << 2;       // float offset, step 4
    const float* gsrc = y + (size_t)sind[r * TOPK + s] * DIM + q;
    float* ldst = &dst[r][q];
#if defined(USE_ASYNC_LDS)
    __builtin_amdgcn_global_load_async_to_lds_b128(
        (GLOBAL_AS v4i*)gsrc, (LDS_AS v4i*)ldst, 0, 0);
#else
    *(float4*)ldst = *(const float4*)gsrc;
#endif
  }
}

__global__ __launch_bounds__(256) void slerp16_kernel(
    const float* __restrict__ y, const int* __restrict__ inds,
    const float* __restrict__ wts, float* __restrict__ out_mu)
{
  const int tid = threadIdx.x;
  const int rbase = blockIdx.x * 16;

  __shared__ float mu[16][DIM];    // 32 KB
  __shared__ float vb[16][DIM];    // 32 KB
  __shared__ float dtile[32 * 8];  // wave 0's 16x16 D tile (1 KB)
  __shared__ int   sind[256];
  __shared__ float swt[256];
  __shared__ float sdot[16], ss1[16], ss2[16], wsum[16];

  sind[tid] = inds[(size_t)rbase * TOPK + tid];
  swt[tid]  = wts[(size_t)rbase * TOPK + tid];
  __syncthreads();

  // mu = y[inds[:,0]]
  loadrows16(y, sind, 0, mu, tid);
#if defined(USE_ASYNC_LDS)
  __builtin_amdgcn_s_wait_asynccnt(0);
#endif
  if (tid < 16) wsum[tid] = swt[tid * TOPK + 0];
  __syncthreads();

#pragma unroll 1
  for (int s = 1; s < TOPK; ++s) {
    loadrows16(y, sind, s, vb, tid);
#if defined(USE_ASYNC_LDS)
    __builtin_amdgcn_s_wait_asynccnt(0);
#endif
    __syncthreads();

    // --- 16 batched dots: diag(mu * v^T) on wave 0 ---
#if defined(USE_WMMA_F32)
    if (tid < 32) {                   // exactly wave 0 => EXEC all ones
      v8f acc0 = {0.f, 0.f, 0.f, 0.f, 0.f, 0.f, 0.f, 0.f};
      v8f acc1 = {0.f, 0.f, 0.f, 0.f, 0.f, 0.f, 0.f, 0.f};
      const int m  = tid & 15;
      const int hi = tid >> 4;
#pragma unroll 1
      for (int k = 0; k < DIM; k += 8) {
        // A (16x4 f32): lanes 0-15 hold K={k,k+1}, lanes 16-31 K={k+2,k+3}
        // B (4x16 f32): same striping with N = lane&15
        const v2f a0 = *(const v2f*)&mu[m][k + 2 * hi];
        const v2f b0 = *(const v2f*)&vb[m][k + 2 * hi];
        acc0 = __builtin_amdgcn_wmma_f32_16x16x4_f32(
            false, a0, false, b0, (short)0, acc0, false, false);
        const v2f a1 = *(const v2f*)&mu[m][k + 4 + 2 * hi];
        const v2f b1 = *(const v2f*)&vb[m][k + 4 + 2 * hi];
        acc1 = __builtin_amdgcn_wmma_f32_16x16x4_f32(
            false, a1, false, b1, (short)0, acc1, false, false);
      }
      const v8f acc = acc0 + acc1;
      // Dump full tile; diag picked out after the barrier.
#pragma unroll
      for (int i = 0; i < 8; ++i) dtile[tid * 8 + i] = acc[i];
    }
#else
    if (tid < 16) {
      float sacc = 0.f;
      for (int dd = 0; dd < DIM; ++dd) sacc += mu[tid][dd] * vb[tid][dd];
      sdot[tid] = sacc;
    }
#endif
    __syncthreads();

    // --- slerp scalars (one thread per row) ---
    if (tid < 16) {
#if defined(USE_WMMA_F32)
      // D[i][i]: i<8 -> lane i, comp i ; i>=8 -> lane i+16, comp i-8
      const int lane = (tid < 8) ? tid : (tid + 16);
      const int comp = (tid < 8) ? tid : (tid - 8);
      const float dotv = dtile[lane * 8 + comp];
#else
      const float dotv = sdot[tid];
#endif
      const float w = swt[tid * TOPK + s];
      const float c = fminf(fmaxf(dotv, -1.0f + 1e-7f), 1.0f - 1e-7f);
      const float theta = acosf(c);
      const float nws = wsum[tid] + w;
      wsum[tid] = nws;
      const float t = w / nws;
      const float st = sinf(theta);
      ss1[tid] = sinf((1.0f - t) * theta) / st;
      ss2[tid] = sinf(t * theta) / st;
    }
    __syncthreads();

    // --- mu update: 16x512 elements, 32 per thread ---
#pragma unroll 4
    for (int j = 0; j < 32; ++j) {
      const int e = tid + 256 * j;
      const int r = e >> 9;
      const int dd = e & (DIM - 1);
      mu[r][dd] = ss1[r] * mu[r][dd] + ss2[r] * vb[r][dd];
    }
    __syncthreads();
  }

  // write mu out (coalesced)
#pragma unroll 4
  for (int j = 0; j < 32; ++j) {
    const int e = tid + 256 * j;
    const int r = e >> 9;
    const int dd = e & (DIM - 1);
    out_mu[(size_t)(rbase + r) * DIM + dd] = mu[r][dd];
  }
}

// ---------------------------------------------------------------------------
extern "C" void kernel_launch(void* const* d_in, const int* in_sizes, int n_in,
                              void* d_out, int out_size, void* d_ws, size_t ws_size,
                              hipStream_t stream) {
  (void)in_sizes; (void)n_in; (void)out_size; (void)ws_size;
  const float* K = (const float*)d_in[0];
  const float* y = (const float*)d_in[1];
  // d_out = [mu: B*D floats][inds: B*TOPK int32]
  float* out_mu  = (float*)d_out;
  int*   out_inds = (int*)(out_mu + (size_t)B_ROWS * DIM);
  float* wts = (float*)d_ws;   // B*TOPK floats of scratch (256 KB)

  topk16_kernel<<<B_ROWS, 256, 0, stream>>>(K, out_inds, wts);
  slerp16_kernel<<<B_ROWS / 16, 256, 0, stream>>>(y, out_inds, wts, out_mu);
}